// AdaptivePersistenceLandscapeLayer_85117661872523
// MI455X (gfx1250) — compile-verified
//
#include <hip/hip_runtime.h>

typedef __attribute__((ext_vector_type(2))) float v2f;
typedef __attribute__((ext_vector_type(8))) float v8f;

#define B_    128
#define N_    65536
#define P_    1024
#define T_    512
#define KMAX  2
#define BIG   1e30f

__global__ __launch_bounds__(256)
void apl_kernel(const float* __restrict__ dtm,
                const int*   __restrict__ birth_loc,
                const int*   __restrict__ death_loc,
                const int*   __restrict__ ph_dim,
                float*       __restrict__ out)
{
    // birthM/deathM interleaved for ds_load_b64; masked-out pairs -> +/-BIG so tent == 0
    __shared__ float2 s_bd[P_];
    __shared__ int    s_mask[P_];
    __shared__ float  s_tmin, s_step;
    __shared__ int    s_has;

    const int blk = blockIdx.x;        // 0..255
    const int b   = blk >> 1;
    const int d   = blk & 1;
    const int tid = threadIdx.x;

    // ---- Phase 1: gather birth/death values, build masked LDS table ----
    for (int p = tid; p < P_; p += 256) {
        int   bl  = birth_loc[b * P_ + p];
        int   dl  = death_loc[b * P_ + p];
        int   dim = ph_dim  [b * P_ + p];
        float bv  = dtm[b * N_ + bl];
        float dv  = dtm[b * N_ + dl];
        int   m   = (dim == d);
        float2 bd;
        bd.x = m ? bv :  BIG;
        bd.y = m ? dv : -BIG;
        s_bd[p]   = bd;
        s_mask[p] = m;
    }
    __syncthreads();

    // ---- Phase 2: count + adaptive window from first min(count,2) masked pairs ----
    if (tid == 0) {
        int count = 0;
        float tmin = BIG, tmax = -BIG;
        for (int p = 0; p < P_; ++p) {
            if (s_mask[p]) {
                if (count < KMAX) {
                    tmin = fminf(tmin, s_bd[p].x);   // birth of early pairs
                    tmax = fmaxf(tmax, s_bd[p].y);   // death of early pairs
                }
                ++count;
            }
        }
        if (count == 0) { tmin = 0.0f; tmax = 1.0f; }
        s_tmin = tmin;
        s_step = (tmax - tmin) / (float)(T_ - 1);
        s_has  = (count > 0);
    }
    __syncthreads();

    const float tmin   = s_tmin;
    const float step   = s_step;
    const int   has    = s_has;

    const int wave   = tid >> 5;      // 0..7 (wave32)
    const int lane   = tid & 31;
    const int col    = lane & 15;     // p-column within tile / t-row for A
    const int halfHi = lane >> 4;     // 0: D rows 0..7 ; 1: D rows 8..15

    // ---- Phase 3: per 16-row t-tile, scan 64 p-tiles with WMMA outer-differences ----
    for (int tile = wave; tile < T_ / 16; tile += 8) {
        const int t_base = tile * 16;

        // A (16x4 f32): row m = [tseq[m], 0.5, tseq[m], 0.5] replicated across K-halves
        v2f afrag;
        afrag.x = tmin + step * (float)(t_base + col);
        afrag.y = 0.5f;

        v8f top1, top2;                     // running top-2 per (row, lane-p-slice)
        #pragma unroll
        for (int r = 0; r < 8; ++r) { top1[r] = 0.0f; top2[r] = 0.0f; }

        for (int p0 = 0; p0 < P_; p0 += 16) {
            float2 bd = s_bd[p0 + col];
            // B1 col n = [0.5, -birth[n], 0.5, -birth[n]]  -> D1 = tseq - birth
            // B2 col n = [-0.5, death[n], -0.5, death[n]]  -> D2 = death - tseq
            v2f b1, b2;
            b1.x =  0.5f;  b1.y = -bd.x;
            b2.x = -0.5f;  b2.y =  bd.y;
            v8f zero = {};
            v8f d1 = __builtin_amdgcn_wmma_f32_16x16x4_f32(
                         false, afrag, false, b1, (short)0, zero, false, false);
            v8f d2 = __builtin_amdgcn_wmma_f32_16x16x4_f32(
                         false, afrag, false, b2, (short)0, zero, false, false);
            #pragma unroll
            for (int r = 0; r < 8; ++r) {
                float tent = fmaxf(fminf(d1[r], d2[r]), 0.0f);
                float mn   = fminf(top1[r], tent);
                top1[r]    = fmaxf(top1[r], tent);
                top2[r]    = fmaxf(top2[r], mn);
            }
        }

        // Butterfly top-2 merge across the 16-lane half (xor masks keep bit4 fixed)
        #pragma unroll
        for (int m = 1; m <= 8; m <<= 1) {
            #pragma unroll
            for (int r = 0; r < 8; ++r) {
                float oa = __shfl_xor(top1[r], m, 32);
                float os = __shfl_xor(top2[r], m, 32);
                float A  = fmaxf(top1[r], oa);
                float S  = fmaxf(fminf(top1[r], oa), fmaxf(top2[r], os));
                top1[r] = A;
                top2[r] = S;
            }
        }

        // Lane 0 writes rows 0..7, lane 16 writes rows 8..15 of this tile
        if (col == 0) {
            const int rowbase = t_base + halfHi * 8;
            float2* outp = (float2*)(out + ((size_t)(b * 2 + d) * T_ + rowbase) * KMAX);
            #pragma unroll
            for (int r = 0; r < 8; ++r) {
                float ts = tmin + step * (float)(rowbase + r);
                float2 res;
                res.x = has ? (top1[r] + ts) : 0.0f;
                res.y = has ? (top2[r] + ts) : 0.0f;
                outp[r] = res;
            }
        }
    }
}

extern "C" void kernel_launch(void* const* d_in, const int* in_sizes, int n_in,
                              void* d_out, int out_size, void* d_ws, size_t ws_size,
                              hipStream_t stream) {
    (void)in_sizes; (void)n_in; (void)d_ws; (void)ws_size; (void)out_size;
    const float* dtm       = (const float*)d_in[0];
    const int*   birth_loc = (const int*)  d_in[1];
    const int*   death_loc = (const int*)  d_in[2];
    const int*   ph_dim    = (const int*)  d_in[3];
    float*       out       = (float*)      d_out;

    dim3 grid(B_ * 2);     // one block per (sample, homology dim)
    dim3 block(256);       // 8 wave32 waves
    apl_kernel<<<grid, block, 0, stream>>>(dtm, birth_loc, death_loc, ph_dim, out);
}